// PureSAGECurvature_14405320311484
// MI455X (gfx1250) — compile-verified
//
#include <hip/hip_runtime.h>
#include <hip/hip_bf16.h>

typedef float v2f __attribute__((ext_vector_type(2)));
typedef float v8f __attribute__((ext_vector_type(8)));

#define LN_EPS 1e-5f

// ---------------------------------------------------------------------------
// Utility: zero a float buffer
// ---------------------------------------------------------------------------
__global__ void fill_zero_kernel(float* __restrict__ p, long long n) {
    long long i = (long long)blockIdx.x * blockDim.x + threadIdx.x;
    if (i < n) p[i] = 0.0f;
}

// ---------------------------------------------------------------------------
// Degree count: one thread per edge, float atomic (exact for deg < 2^24)
// ---------------------------------------------------------------------------
__global__ void count_deg_kernel(const int* __restrict__ dst, float* __restrict__ deg, int E) {
    int e = blockIdx.x * blockDim.x + threadIdx.x;
    if (e < E) atomicAdd(&deg[dst[e]], 1.0f);
}

__global__ void inv_deg_kernel(float* __restrict__ deg, int N) {
    int i = blockIdx.x * blockDim.x + threadIdx.x;
    if (i < N) deg[i] = 1.0f / fmaxf(deg[i], 1.0f);
}

// ---------------------------------------------------------------------------
// Edge scatter-add: thread = (edge, 4-float chunk). Gather is a coalesced
// global_load_b128 per lane; adds land in L2 (h/agg are L2-resident: 51MB).
// ---------------------------------------------------------------------------
__global__ void scatter_add_kernel(const float* __restrict__ h,
                                   const int* __restrict__ src,
                                   const int* __restrict__ dst,
                                   float* __restrict__ agg,
                                   long long total, int D) {
    long long tid = (long long)blockIdx.x * blockDim.x + threadIdx.x;
    if (tid >= total) return;
    const int chunks = D >> 2;
    const int e = (int)(tid / chunks);
    const int c = (int)(tid % chunks) << 2;
    const int s = src[e];
    const int d = dst[e];
    const float4 v = *(const float4*)(h + (long long)s * D + c);
    float* o = agg + (long long)d * D + c;
    atomicAdd(o + 0, v.x);
    atomicAdd(o + 1, v.y);
    atomicAdd(o + 2, v.z);
    atomicAdd(o + 3, v.w);
}

// ---------------------------------------------------------------------------
// Fused SAGE layer (K=128): out = relu(LN(agg/deg @ Wl + b + hin @ Wr)) + hin
// One wave owns 16 rows x 128 cols: 8 f32 16x16 accumulator tiles.
// A 16x4 fragment: lanes 0-15 hold K0/K1, lanes 16-31 hold K2/K3 (ISA layout).
// LayerNorm reduced with 16-lane shfl_xor inside the wave. Safe in-place.
// ---------------------------------------------------------------------------
__global__ __launch_bounds__(256)
void fused_sage128_kernel(const float* __restrict__ agg,
                          const float* __restrict__ inv_deg,
                          const float* __restrict__ hin,
                          const float* __restrict__ Wl,   // 128x128 row-major (K x out)
                          const float* __restrict__ bl,   // 128
                          const float* __restrict__ Wr,   // 128x128
                          const float* __restrict__ g,    // 128
                          const float* __restrict__ be,   // 128
                          float* __restrict__ hout,
                          int N) {
    const int wave = blockIdx.x * (blockDim.x >> 5) + (threadIdx.x >> 5);
    const int lane = threadIdx.x & 31;
    const int row0 = wave << 4;
    if (row0 >= N) return;

    const int rl = lane & 15;           // A: row-in-tile | B: col-in-tile
    const int kh = (lane >> 4) << 1;    // K sub-offset per half-wave
    const int rowc = min(row0 + rl, N - 1);
    const long long arow = (long long)rowc * 128;
    const float scale = inv_deg[rowc];

    v8f acc[8] = {};
    for (int k = 0; k < 128; k += 4) {
        const int kk = k + kh;
        v2f aA, aH;
        aA.x = agg[arow + kk] * scale;
        aA.y = agg[arow + kk + 1] * scale;
        aH.x = hin[arow + kk];
        aH.y = hin[arow + kk + 1];
#pragma unroll
        for (int t = 0; t < 8; ++t) {
            const int col = (t << 4) + rl;
            v2f bL, bR;
            bL.x = Wl[kk * 128 + col];
            bL.y = Wl[(kk + 1) * 128 + col];
            bR.x = Wr[kk * 128 + col];
            bR.y = Wr[(kk + 1) * 128 + col];
            acc[t] = __builtin_amdgcn_wmma_f32_16x16x4_f32(false, aA, false, bL,
                                                           (short)0, acc[t], false, false);
            acc[t] = __builtin_amdgcn_wmma_f32_16x16x4_f32(false, aH, false, bR,
                                                           (short)0, acc[t], false, false);
        }
    }
    // bias (per output column)
#pragma unroll
    for (int t = 0; t < 8; ++t) {
        const float b = bl[(t << 4) + rl];
#pragma unroll
        for (int r = 0; r < 8; ++r) acc[t][r] += b;
    }
    // LayerNorm stats per row (16-lane reduction within each half-wave)
    float mu[8], rn[8];
#pragma unroll
    for (int r = 0; r < 8; ++r) {
        float s = 0.0f, q = 0.0f;
#pragma unroll
        for (int t = 0; t < 8; ++t) { const float v = acc[t][r]; s += v; q += v * v; }
#pragma unroll
        for (int m = 8; m >= 1; m >>= 1) {
            s += __shfl_xor(s, m, 32);
            q += __shfl_xor(q, m, 32);
        }
        const float m1 = s * (1.0f / 128.0f);
        const float var = q * (1.0f / 128.0f) - m1 * m1;
        mu[r] = m1;
        rn[r] = rsqrtf(var + LN_EPS);
    }
    // affine + relu + residual + store
    const int rbase = row0 + ((lane >> 4) << 3);
#pragma unroll
    for (int t = 0; t < 8; ++t) {
        const int col = (t << 4) + rl;
        const float gg = g[col], bb = be[col];
#pragma unroll
        for (int r = 0; r < 8; ++r) {
            const int rowm = rbase + r;
            if (rowm < N) {
                float v = (acc[t][r] - mu[r]) * rn[r] * gg + bb;
                v = fmaxf(v, 0.0f);
                const long long idx = (long long)rowm * 128 + col;
                hout[idx] = v + hin[idx];
            }
        }
    }
}

// ---------------------------------------------------------------------------
// Fused layer 0 (K=32): out = relu(LN(agg/deg @ Wl0 + b + x @ Wr0)) + x @ Wproj
// Second accumulator set folds the input-projection GEMM in (no extra buffer).
// ---------------------------------------------------------------------------
__global__ __launch_bounds__(256)
void fused_sage0_kernel(const float* __restrict__ agg,   // N x 32
                        const float* __restrict__ inv_deg,
                        const float* __restrict__ x,     // N x 32
                        const float* __restrict__ Wl,    // 32x128
                        const float* __restrict__ bl,
                        const float* __restrict__ Wr,    // 32x128
                        const float* __restrict__ Wp,    // 32x128
                        const float* __restrict__ g,
                        const float* __restrict__ be,
                        float* __restrict__ hout,        // N x 128
                        int N) {
    const int wave = blockIdx.x * (blockDim.x >> 5) + (threadIdx.x >> 5);
    const int lane = threadIdx.x & 31;
    const int row0 = wave << 4;
    if (row0 >= N) return;

    const int rl = lane & 15;
    const int kh = (lane >> 4) << 1;
    const int rowc = min(row0 + rl, N - 1);
    const long long arow = (long long)rowc * 32;
    const float scale = inv_deg[rowc];

    v8f acc[8] = {};
    v8f prj[8] = {};
    for (int k = 0; k < 32; k += 4) {
        const int kk = k + kh;
        v2f aA, aX;
        aA.x = agg[arow + kk] * scale;
        aA.y = agg[arow + kk + 1] * scale;
        aX.x = x[arow + kk];
        aX.y = x[arow + kk + 1];
#pragma unroll
        for (int t = 0; t < 8; ++t) {
            const int col = (t << 4) + rl;
            v2f bL, bR, bP;
            bL.x = Wl[kk * 128 + col];
            bL.y = Wl[(kk + 1) * 128 + col];
            bR.x = Wr[kk * 128 + col];
            bR.y = Wr[(kk + 1) * 128 + col];
            bP.x = Wp[kk * 128 + col];
            bP.y = Wp[(kk + 1) * 128 + col];
            acc[t] = __builtin_amdgcn_wmma_f32_16x16x4_f32(false, aA, false, bL,
                                                           (short)0, acc[t], false, false);
            acc[t] = __builtin_amdgcn_wmma_f32_16x16x4_f32(false, aX, false, bR,
                                                           (short)0, acc[t], false, false);
            prj[t] = __builtin_amdgcn_wmma_f32_16x16x4_f32(false, aX, false, bP,
                                                           (short)0, prj[t], false, false);
        }
    }
#pragma unroll
    for (int t = 0; t < 8; ++t) {
        const float b = bl[(t << 4) + rl];
#pragma unroll
        for (int r = 0; r < 8; ++r) acc[t][r] += b;
    }
    float mu[8], rn[8];
#pragma unroll
    for (int r = 0; r < 8; ++r) {
        float s = 0.0f, q = 0.0f;
#pragma unroll
        for (int t = 0; t < 8; ++t) { const float v = acc[t][r]; s += v; q += v * v; }
#pragma unroll
        for (int m = 8; m >= 1; m >>= 1) {
            s += __shfl_xor(s, m, 32);
            q += __shfl_xor(q, m, 32);
        }
        const float m1 = s * (1.0f / 128.0f);
        const float var = q * (1.0f / 128.0f) - m1 * m1;
        mu[r] = m1;
        rn[r] = rsqrtf(var + LN_EPS);
    }
    const int rbase = row0 + ((lane >> 4) << 3);
#pragma unroll
    for (int t = 0; t < 8; ++t) {
        const int col = (t << 4) + rl;
        const float gg = g[col], bb = be[col];
#pragma unroll
        for (int r = 0; r < 8; ++r) {
            const int rowm = rbase + r;
            if (rowm < N) {
                float v = (acc[t][r] - mu[r]) * rn[r] * gg + bb;
                v = fmaxf(v, 0.0f);
                hout[(long long)rowm * 128 + col] = v + prj[t][r];
            }
        }
    }
}

// ---------------------------------------------------------------------------
// Head: yhat[n] = h[n,:] . Wh + bh  (one wave per node, coalesced float4)
// ---------------------------------------------------------------------------
__global__ void head_kernel(const float* __restrict__ h,
                            const float* __restrict__ Wh,
                            const float* __restrict__ bh,
                            float* __restrict__ y, int N) {
    const int wave = blockIdx.x * (blockDim.x >> 5) + (threadIdx.x >> 5);
    const int lane = threadIdx.x & 31;
    if (wave >= N) return;
    const float4 a = *(const float4*)(h + (long long)wave * 128 + (lane << 2));
    const float4 w = *(const float4*)(Wh + (lane << 2));
    float s = a.x * w.x + a.y * w.y + a.z * w.z + a.w * w.w;
#pragma unroll
    for (int m = 16; m >= 1; m >>= 1) s += __shfl_xor(s, m, 32);
    if (lane == 0) y[wave] = s + bh[0];
}

// ---------------------------------------------------------------------------
// Host orchestration
// ---------------------------------------------------------------------------
extern "C" void kernel_launch(void* const* d_in, const int* in_sizes, int n_in,
                              void* d_out, int out_size, void* d_ws, size_t ws_size,
                              hipStream_t stream) {
    const float* x     = (const float*)d_in[0];
    const int*   ei    = (const int*)d_in[1];
    const float* Wl0   = (const float*)d_in[2];
    const float* bl0   = (const float*)d_in[3];
    const float* Wr0   = (const float*)d_in[4];
    const float* g0    = (const float*)d_in[5];
    const float* be0   = (const float*)d_in[6];
    const float* Wl1   = (const float*)d_in[7];
    const float* bl1   = (const float*)d_in[8];
    const float* Wr1   = (const float*)d_in[9];
    const float* g1    = (const float*)d_in[10];
    const float* be1   = (const float*)d_in[11];
    const float* Wl2   = (const float*)d_in[12];
    const float* bl2   = (const float*)d_in[13];
    const float* Wr2   = (const float*)d_in[14];
    const float* g2    = (const float*)d_in[15];
    const float* be2   = (const float*)d_in[16];
    const float* Wproj = (const float*)d_in[17];
    const float* Wh    = (const float*)d_in[18];
    const float* bh    = (const float*)d_in[19];

    const int N = in_sizes[0] / 32;
    const int E = in_sizes[1] / 2;
    const int* src = ei;
    const int* dst = ei + E;

    // workspace: [deg|inv_deg : N floats][agg : N*128 floats]
    float* deg = (float*)d_ws;
    float* agg = deg + N;

    // output: [yhat : N][h : N*128] ; layers 1/2 run in-place on the h region
    float* yhat = (float*)d_out;
    float* Hbuf = yhat + N;

    const int TPB = 256;
    const int waves16 = (N + 15) / 16;
    const int fusedBlocks = (waves16 + 7) / 8;

    // 1. degree -> inv_deg
    fill_zero_kernel<<<(N + TPB - 1) / TPB, TPB, 0, stream>>>(deg, (long long)N);
    count_deg_kernel<<<(E + TPB - 1) / TPB, TPB, 0, stream>>>(dst, deg, E);
    inv_deg_kernel<<<(N + TPB - 1) / TPB, TPB, 0, stream>>>(deg, N);

    // 2. layer 0 aggregation (D=32)
    {
        const long long nz = (long long)N * 32;
        fill_zero_kernel<<<(int)((nz + TPB - 1) / TPB), TPB, 0, stream>>>(agg, nz);
        const long long total = (long long)E * 8;
        scatter_add_kernel<<<(int)((total + TPB - 1) / TPB), TPB, 0, stream>>>(
            x, src, dst, agg, total, 32);
    }
    fused_sage0_kernel<<<fusedBlocks, TPB, 0, stream>>>(
        agg, deg, x, Wl0, bl0, Wr0, Wproj, g0, be0, Hbuf, N);

    // 3. layer 1 (D=128, in-place)
    {
        const long long nz = (long long)N * 128;
        fill_zero_kernel<<<(int)((nz + TPB - 1) / TPB), TPB, 0, stream>>>(agg, nz);
        const long long total = (long long)E * 32;
        scatter_add_kernel<<<(int)((total + TPB - 1) / TPB), TPB, 0, stream>>>(
            Hbuf, src, dst, agg, total, 128);
    }
    fused_sage128_kernel<<<fusedBlocks, TPB, 0, stream>>>(
        agg, deg, Hbuf, Wl1, bl1, Wr1, g1, be1, Hbuf, N);

    // 4. layer 2 (D=128, in-place)
    {
        const long long nz = (long long)N * 128;
        fill_zero_kernel<<<(int)((nz + TPB - 1) / TPB), TPB, 0, stream>>>(agg, nz);
        const long long total = (long long)E * 32;
        scatter_add_kernel<<<(int)((total + TPB - 1) / TPB), TPB, 0, stream>>>(
            Hbuf, src, dst, agg, total, 128);
    }
    fused_sage128_kernel<<<fusedBlocks, TPB, 0, stream>>>(
        agg, deg, Hbuf, Wl2, bl2, Wr2, g2, be2, Hbuf, N);

    // 5. head
    head_kernel<<<(N + 7) / 8, TPB, 0, stream>>>(Hbuf, Wh, bh, yhat, N);
}